// MADG_48533130445499
// MI455X (gfx1250) — compile-verified
//
#include <hip/hip_runtime.h>

#define S      8
#define PAD    10000
#define NA     2500
#define NM     500
#define MK     3
#define FEAT   64
#define HID    64
#define NH     8
#define NTOT   (NA + NM)

typedef __attribute__((ext_vector_type(2))) float v2f;
typedef __attribute__((ext_vector_type(8))) float v8f;

// ---------------------------------------------------------------- utilities
__global__ void zero_kernel(float* __restrict__ p, long n) {
    long t = (long)blockIdx.x * blockDim.x + threadIdx.x;
    if (t < n) p[t] = 0.0f;
}

// dst[s*dstStride + i*64 + d] = src[s*srcStride + idx[s*NA+i]*64 + d]
__global__ void gather_rows_kernel(float* __restrict__ dst, long dstStride,
                                   const float* __restrict__ src, long srcStride,
                                   const int* __restrict__ idx) {
    long t = (long)blockIdx.x * blockDim.x + threadIdx.x;
    long total = (long)S * NA * 64;
    if (t >= total) return;
    int d = (int)(t & 63);
    long r = t >> 6;
    int i = (int)(r % NA);
    int s = (int)(r / NA);
    int node = idx[(long)s * NA + i];
    dst[(long)s * dstStride + (long)i * 64 + d] =
        src[(long)s * srcStride + (long)node * 64 + d];
}

// X[s][nodes[s][i]][d] = H[s][i][d]
__global__ void scatter_rows_kernel(float* __restrict__ X,
                                    const float* __restrict__ H,
                                    const int* __restrict__ idx) {
    long t = (long)blockIdx.x * blockDim.x + threadIdx.x;
    long total = (long)S * NA * 64;
    if (t >= total) return;
    int d = (int)(t & 63);
    long r = t >> 6;
    int i = (int)(r % NA);
    int s = (int)(r / NA);
    int node = idx[(long)s * NA + i];
    X[((long)s * PAD + node) * 64 + d] = H[((long)s * NTOT + i) * 64 + d];
}

// ------------------------------------------------------ motif feature extract
// One 64-thread block per motif. Only the learned token's output row is needed:
// q is computed for the token only; k,v for token + 3 member rows.
__global__ void motif_kernel(const float* __restrict__ rows, long rowStride,
                             const int* __restrict__ motifs,   // [S][NM][3]
                             const float* __restrict__ token,  // [64]
                             const float* __restrict__ W,      // [4][64][64]
                             const float* __restrict__ b,      // [4][64]
                             float* __restrict__ outRows, long outStride,
                             int outRowOffset) {
    int m = blockIdx.x;            // motif
    int s = blockIdx.y;            // snapshot
    int d = threadIdx.x;           // 0..63 (output dim / column)

    __shared__ float xr[4][64];
    __shared__ float q0[64], kk[4][64], vv[4][64];
    __shared__ float sc[8][4], aw[8][4];
    __shared__ float ob[64];

    xr[0][d] = token[d];
    const float* base = rows + (long)s * rowStride;
    #pragma unroll
    for (int j = 0; j < MK; ++j) {
        int idx = motifs[((long)s * NM + m) * MK + j];
        xr[j + 1][d] = base[(long)idx * 64 + d];
    }
    __syncthreads();

    const float* Wq = W;
    const float* Wk = W + 64 * 64;
    const float* Wv = W + 2 * 64 * 64;
    const float* Wo = W + 3 * 64 * 64;

    float accq = b[d];
    float acck[4] = {b[64 + d], b[64 + d], b[64 + d], b[64 + d]};
    float accv[4] = {b[128 + d], b[128 + d], b[128 + d], b[128 + d]};
    for (int k = 0; k < 64; ++k) {
        float wq = Wq[k * 64 + d];
        float wk = Wk[k * 64 + d];
        float wv = Wv[k * 64 + d];
        accq += xr[0][k] * wq;
        #pragma unroll
        for (int j = 0; j < 4; ++j) {
            acck[j] += xr[j][k] * wk;
            accv[j] += xr[j][k] * wv;
        }
    }
    q0[d] = accq;
    #pragma unroll
    for (int j = 0; j < 4; ++j) { kk[j][d] = acck[j]; vv[j][d] = accv[j]; }
    __syncthreads();

    if (d < 32) {                      // scores: h = d>>2, j = d&3
        int h = d >> 2, j = d & 3;
        float sacc = 0.0f;
        #pragma unroll
        for (int e = 0; e < 8; ++e) sacc += q0[h * 8 + e] * kk[j][h * 8 + e];
        sc[h][j] = sacc * 0.35355339059327373f;   // 1/sqrt(8)
    }
    __syncthreads();
    if (d < 32) {                      // softmax over j
        int h = d >> 2, j = d & 3;
        float mx = fmaxf(fmaxf(sc[h][0], sc[h][1]), fmaxf(sc[h][2], sc[h][3]));
        float e0 = __expf(sc[h][0] - mx), e1 = __expf(sc[h][1] - mx);
        float e2 = __expf(sc[h][2] - mx), e3 = __expf(sc[h][3] - mx);
        aw[h][j] = __expf(sc[h][j] - mx) / (e0 + e1 + e2 + e3);
    }
    __syncthreads();
    {                                  // weighted value sum
        int h = d >> 3;
        float o = 0.0f;
        #pragma unroll
        for (int j = 0; j < 4; ++j) o += aw[h][j] * vv[j][d];
        ob[d] = o;
    }
    __syncthreads();
    {                                  // output projection
        float acc = b[192 + d];
        for (int k = 0; k < 64; ++k) acc += ob[k] * Wo[k * 64 + d];
        outRows[(long)s * outStride + (long)(outRowOffset + m) * 64 + d] = acc;
    }
}

// --------------------------------------------------------- WMMA GEMM (N==64)
// C[M,64] = A[M,K] @ B[K,64] (+bias)(+relu), batched via grid.y and strides.
// One wave computes a 16x64 output strip (4 accumulators, A-frag reused).
__global__ void wmma_gemm_n64(const float* __restrict__ A, long lda, long strideA,
                              const float* __restrict__ B, long strideB,
                              float* __restrict__ C, long strideC,
                              int M, int K,
                              const float* __restrict__ bias, int relu) {
    int s = blockIdx.y;
    const float* Ab = A + (long)s * strideA;
    const float* Bb = B + (long)s * strideB;
    float*       Cb = C + (long)s * strideC;

    int wave = threadIdx.x >> 5;
    int lane = threadIdx.x & 31;
    int mt = blockIdx.x * 4 + wave;      // 16-row tile index
    int m0 = mt * 16;
    if (m0 >= M) return;

    // A fragment (16x4 f32): lanes 0-15 -> M rows, K in {0,1}; lanes 16-31 -> K in {2,3}
    int am = lane & 15;
    int ak = (lane >> 4) * 2;
    int arow = m0 + am; if (arow >= M) arow = M - 1;
    // B fragment (4x16): lanes 0-15 -> K row bk, lanes 16-31 -> K row bk+?
    int bn = lane & 15;
    int bk = (lane >> 4) * 2;

    v8f acc[4] = {};
    for (int k0 = 0; k0 < K; k0 += 4) {
        v2f a;
        a.x = Ab[(long)arow * lda + k0 + ak];
        a.y = Ab[(long)arow * lda + k0 + ak + 1];
        #pragma unroll
        for (int nt = 0; nt < 4; ++nt) {
            v2f bfrag;
            bfrag.x = Bb[(long)(k0 + bk) * 64 + nt * 16 + bn];
            bfrag.y = Bb[(long)(k0 + bk + 1) * 64 + nt * 16 + bn];
            acc[nt] = __builtin_amdgcn_wmma_f32_16x16x4_f32(
                false, a, false, bfrag, (short)0, acc[nt], false, false);
        }
    }

    int cn = lane & 15;
    int rbase = m0 + (lane >> 4) * 8;
    #pragma unroll
    for (int nt = 0; nt < 4; ++nt) {
        #pragma unroll
        for (int r = 0; r < 8; ++r) {
            int row = rbase + r;
            if (row < M) {
                float v = acc[nt][r];
                int col = nt * 16 + cn;
                if (bias) v += bias[col];
                if (relu) v = fmaxf(v, 0.0f);
                Cb[(long)row * 64 + col] = v;
            }
        }
    }
}

// ----------------------------------------------------------- Z @ Z^T decoder
__global__ void zzt_kernel(const float* __restrict__ Z, float* __restrict__ Out) {
    int s = blockIdx.z;
    const float* Zb = Z + (long)s * NTOT * 64;
    float*       Ob = Out + (long)s * NTOT * NTOT;

    int wave = threadIdx.x >> 5;
    int lane = threadIdx.x & 31;
    int nt = blockIdx.x * 4 + wave;      // 16-col tile
    int mt = blockIdx.y;                 // 16-row tile
    int m0 = mt * 16, n0 = nt * 16;
    if (n0 >= NTOT || m0 >= NTOT) return;

    int am = lane & 15, ak = (lane >> 4) * 2;
    int bn = lane & 15, bk = (lane >> 4) * 2;
    int arow = m0 + am; if (arow >= NTOT) arow = NTOT - 1;
    int bcol = n0 + bn; if (bcol >= NTOT) bcol = NTOT - 1;

    v8f acc = {};
    #pragma unroll
    for (int k0 = 0; k0 < 64; k0 += 4) {
        v2f a, b;
        a.x = Zb[(long)arow * 64 + k0 + ak];
        a.y = Zb[(long)arow * 64 + k0 + ak + 1];
        b.x = Zb[(long)bcol * 64 + k0 + bk];      // B[k][n] = Z[n][k]
        b.y = Zb[(long)bcol * 64 + k0 + bk + 1];
        acc = __builtin_amdgcn_wmma_f32_16x16x4_f32(
            false, a, false, b, (short)0, acc, false, false);
    }

    int cn = lane & 15;
    int rbase = m0 + (lane >> 4) * 8;
    #pragma unroll
    for (int r = 0; r < 8; ++r) {
        int row = rbase + r;
        int col = n0 + cn;
        if (row < NTOT && col < NTOT)
            Ob[(long)row * NTOT + col] = acc[r];
    }
}

// ------------------------------------------------- temporal attention (L==8)
// One 64-thread block per node; projections were done by WMMA GEMM into Qx/Kx/Vx
// (rows in X order, row index = l*PAD + node). Output projection fused.
__global__ void temporal_attn_kernel(const float* __restrict__ Qx,
                                     const float* __restrict__ Kx,
                                     const float* __restrict__ Vx,
                                     const float* __restrict__ time_emb, // [2S-1][NH]
                                     const float* __restrict__ W,        // mhat_W [4][64][64]
                                     const float* __restrict__ bfull,    // mhat_b [4][64]
                                     float* __restrict__ XTb) {          // [S][PAD][64]
    int node = blockIdx.x;
    int d = threadIdx.x;

    __shared__ float qb[8][64], kb[8][64], vb[8][64];
    __shared__ float ab[8][8][8];   // [h][l][m]
    __shared__ float obuf[8][64];

    #pragma unroll
    for (int l = 0; l < S; ++l) {
        long r = ((long)l * PAD + node) * 64 + d;
        qb[l][d] = Qx[r];
        kb[l][d] = Kx[r];
        vb[l][d] = Vx[r];
    }
    __syncthreads();

    {   // scores + softmax: h = d>>3, l = d&7; each thread handles one (h,l) row
        int h = d >> 3, l = d & 7;
        float srow[8];
        float mx = -3.402823466e38f;
        #pragma unroll
        for (int m2 = 0; m2 < 8; ++m2) {
            float sacc = 0.0f;
            #pragma unroll
            for (int e = 0; e < 8; ++e) sacc += qb[l][h * 8 + e] * kb[m2][h * 8 + e];
            sacc = sacc * 0.35355339059327373f + time_emb[(m2 - l + S - 1) * NH + h];
            srow[m2] = sacc;
            mx = fmaxf(mx, sacc);
        }
        float sum = 0.0f;
        #pragma unroll
        for (int m2 = 0; m2 < 8; ++m2) { srow[m2] = __expf(srow[m2] - mx); sum += srow[m2]; }
        float inv = 1.0f / sum;
        #pragma unroll
        for (int m2 = 0; m2 < 8; ++m2) ab[h][l][m2] = srow[m2] * inv;
    }
    __syncthreads();

    {   // o[l][d] = sum_m a[h][l][m] * v[m][d]
        int h = d >> 3;
        #pragma unroll
        for (int l = 0; l < 8; ++l) {
            float o = 0.0f;
            #pragma unroll
            for (int m2 = 0; m2 < 8; ++m2) o += ab[h][l][m2] * vb[m2][d];
            obuf[l][d] = o;
        }
    }
    __syncthreads();

    {   // output projection, column d for all l
        const float* Wo = W + 3 * 64 * 64;
        const float* bo = bfull + 3 * 64;
        float out[8];
        #pragma unroll
        for (int l = 0; l < 8; ++l) out[l] = bo[d];
        for (int k = 0; k < 64; ++k) {
            float w = Wo[k * 64 + d];
            #pragma unroll
            for (int l = 0; l < 8; ++l) out[l] += obuf[l][k] * w;
        }
        #pragma unroll
        for (int l = 0; l < 8; ++l)
            XTb[((long)l * PAD + node) * 64 + d] = out[l];
    }
}

// ---------------------------------------------------------------- launcher
extern "C" void kernel_launch(void* const* d_in, const int* in_sizes, int n_in,
                              void* d_out, int out_size, void* d_ws, size_t ws_size,
                              hipStream_t stream) {
    (void)in_sizes; (void)n_in; (void)out_size; (void)ws_size;

    const float* node_emb   = (const float*)d_in[0];
    const float* mfe1_token = (const float*)d_in[1];
    const float* mha1_W     = (const float*)d_in[2];
    const float* mha1_b     = (const float*)d_in[3];
    const float* gc_W1      = (const float*)d_in[4];
    const float* gc_b1      = (const float*)d_in[5];
    const float* gc_W2      = (const float*)d_in[6];
    const float* gc_b2      = (const float*)d_in[7];
    const float* time_emb   = (const float*)d_in[8];
    const float* mhat_W     = (const float*)d_in[9];
    const float* mhat_b     = (const float*)d_in[10];
    const float* mfe2_token = (const float*)d_in[11];
    const float* mha2_W     = (const float*)d_in[12];
    const float* mha2_b     = (const float*)d_in[13];
    const float* adj        = (const float*)d_in[14];
    const int*   nodes      = (const int*)d_in[15];
    const int*   motifs     = (const int*)d_in[16];
    float*       out        = (float*)d_out;

    // workspace layout (floats)
    float* ws = (float*)d_ws;
    const long SZ_H  = (long)S * NTOT * 64;   // 1,536,000
    const long SZ_X  = (long)S * PAD * 64;    // 5,120,000
    float* H   = ws;                  // [S][NTOT][64]
    float* G   = H   + SZ_H;          // [S][NTOT][64]
    float* X   = G   + SZ_H;          // [S][PAD][64]
    float* Qx  = X   + SZ_X;          // [S][PAD][64]
    float* Kx  = Qx  + SZ_X;
    float* Vx  = Kx  + SZ_X;
    float* XTb = Vx  + SZ_X;          // [S][PAD][64]
    float* Z   = XTb + SZ_X;          // [S][NTOT][64]

    const int rowTilesNTOT = (NTOT + 15) / 16;           // 188
    dim3 gemmBlk(128, 1, 1);                             // 4 waves
    dim3 gemmGridS((rowTilesNTOT + 3) / 4, S, 1);        // per-snapshot batched

    // 1. zero padded buffer X
    {
        long n = SZ_X;
        zero_kernel<<<(unsigned)((n + 255) / 256), 256, 0, stream>>>(X, n);
    }
    // 2. gather node embeddings into H[s][0:NA]
    {
        long n = (long)S * NA * 64;
        gather_rows_kernel<<<(unsigned)((n + 255) / 256), 256, 0, stream>>>(
            H, (long)NTOT * 64, node_emb, 0L, nodes);
    }
    // 3. encoder motif features into H[s][NA:NTOT]
    motif_kernel<<<dim3(NM, S, 1), 64, 0, stream>>>(
        H, (long)NTOT * 64, motifs, mfe1_token, mha1_W, mha1_b,
        H, (long)NTOT * 64, NA);
    // 4-7. two GCN layers (WMMA)
    wmma_gemm_n64<<<gemmGridS, gemmBlk, 0, stream>>>(
        H, 64L, (long)NTOT * 64, gc_W1, 0L, G, (long)NTOT * 64,
        NTOT, 64, nullptr, 0);
    wmma_gemm_n64<<<gemmGridS, gemmBlk, 0, stream>>>(
        adj, (long)NTOT, (long)NTOT * NTOT, G, (long)NTOT * 64,
        H, (long)NTOT * 64, NTOT, NTOT, gc_b1, 1);
    wmma_gemm_n64<<<gemmGridS, gemmBlk, 0, stream>>>(
        H, 64L, (long)NTOT * 64, gc_W2, 0L, G, (long)NTOT * 64,
        NTOT, 64, nullptr, 0);
    wmma_gemm_n64<<<gemmGridS, gemmBlk, 0, stream>>>(
        adj, (long)NTOT, (long)NTOT * NTOT, G, (long)NTOT * 64,
        H, (long)NTOT * 64, NTOT, NTOT, gc_b2, 1);
    // 8. scatter active rows into padded X
    {
        long n = (long)S * NA * 64;
        scatter_rows_kernel<<<(unsigned)((n + 255) / 256), 256, 0, stream>>>(X, H, nodes);
    }
    // 9. temporal q/k/v projections over all S*PAD rows (WMMA)
    {
        const int Mrows = S * PAD;                        // 80000
        dim3 grid((Mrows / 16 + 3) / 4, 1, 1);            // 1250 blocks
        wmma_gemm_n64<<<grid, gemmBlk, 0, stream>>>(
            X, 64L, 0L, mhat_W + 0 * 64 * 64, 0L, Qx, 0L, Mrows, 64, mhat_b + 0 * 64, 0);
        wmma_gemm_n64<<<grid, gemmBlk, 0, stream>>>(
            X, 64L, 0L, mhat_W + 1 * 64 * 64, 0L, Kx, 0L, Mrows, 64, mhat_b + 1 * 64, 0);
        wmma_gemm_n64<<<grid, gemmBlk, 0, stream>>>(
            X, 64L, 0L, mhat_W + 2 * 64 * 64, 0L, Vx, 0L, Mrows, 64, mhat_b + 2 * 64, 0);
    }
    // 10. per-node temporal attention + fused output projection
    temporal_attn_kernel<<<dim3(PAD, 1, 1), 64, 0, stream>>>(
        Qx, Kx, Vx, time_emb, mhat_W, mhat_b, XTb);
    // 11. decoder gather into Z[s][0:NA]
    {
        long n = (long)S * NA * 64;
        gather_rows_kernel<<<(unsigned)((n + 255) / 256), 256, 0, stream>>>(
            Z, (long)NTOT * 64, XTb, (long)PAD * 64, nodes);
    }
    // 12. decoder motif features into Z[s][NA:NTOT]
    motif_kernel<<<dim3(NM, S, 1), 64, 0, stream>>>(
        Z, (long)NTOT * 64, motifs, mfe2_token, mha2_W, mha2_b,
        Z, (long)NTOT * 64, NA);
    // 13. structure decoder: out[s] = Z[s] @ Z[s]^T (WMMA)
    {
        dim3 grid((rowTilesNTOT + 3) / 4, rowTilesNTOT, S);
        zzt_kernel<<<grid, gemmBlk, 0, stream>>>(Z, out);
    }
}